// LenghtRegulator_80711025427128
// MI455X (gfx1250) — compile-verified
//
#include <hip/hip_runtime.h>
#include <stdint.h>

// LengthRegulator for MI455X (gfx1250).
// B=32, L=256, D=512, MAX_LEN=2048. Output (B, MAX_LEN, D) f32 = 128 MiB.
// Pure data movement: CDNA5 async global->LDS->global pipeline (ASYNCcnt)
// for the gather-copy; NT stores keep the output stream from evicting the
// L2-resident att_out gather set (16 MiB << 192 MB L2).

#define BB      32
#define LL      256
#define DD      512
#define MAXLEN  2048
#define ROWS_PB 8            // one wave32 per output row, 8 waves / block

typedef __attribute__((address_space(3))) float lds_float_t;
typedef float v4f __attribute__((ext_vector_type(4)));

static __device__ __forceinline__ uint32_t lds_byte_offset(float* p) {
  // generic -> addrspace(3); ptrtoint of an AS(3) pointer is the 32-bit LDS
  // byte offset needed by the async ops' VDST/VSRC operand.
  return (uint32_t)(uintptr_t)(lds_float_t*)p;
}

__global__ __launch_bounds__(256) void length_regulator_kernel(
    const float* __restrict__ att,    // (B, L, D)
    const float* __restrict__ dur,    // (B, L)
    const int*   __restrict__ alpha,  // scalar
    float*       __restrict__ out)    // (B, MAX_LEN, D)
{
  __shared__ int   scsum[LL];           // inclusive cumsum of rounded durations
  __shared__ int   wsum[ROWS_PB];       // per-wave partial sums
  __shared__ float xbuf[ROWS_PB][DD];   // 16 KB row-staging for async copies

  const int t    = threadIdx.x;
  const int wave = t >> 5;
  const int lane = t & 31;
  const int row0 = blockIdx.x * ROWS_PB;       // first output row of block
  const int b    = row0 >> 11;                 // row0 / MAX_LEN (8 | 2048)
  const float alphaf = (float)alpha[0];

  // ---- r = rint(duration*alpha), wave32 inclusive scan (no barriers)
  int v = (int)__builtin_rintf(dur[b * LL + t] * alphaf);
#pragma unroll
  for (int d = 1; d < 32; d <<= 1) {
    int n = __shfl_up(v, d, 32);
    if (lane >= d) v += n;
  }
  if (lane == 31) wsum[wave] = v;            // wave totals
  __syncthreads();

  // cross-wave exclusive offset + grand total from the 8 partials
  int off8 = 0, total = 0;
#pragma unroll
  for (int w = 0; w < ROWS_PB; ++w) {
    int s = wsum[w];
    total += s;
    if (w < wave) off8 += s;
  }
  scsum[t] = v + off8;
  __syncthreads();

  const int p = (row0 & (MAXLEN - 1)) + wave;   // output position, wave-uniform
  const uint32_t dstOff =
      ((uint32_t)(b * MAXLEN + p) * DD) * 4u + (uint32_t)lane * 16u;

  if (p < total) {
    // searchsorted(csum, p, side='right') == upper_bound (uniform per wave,
    // LDS broadcast reads)
    int lo = 0, hi = LL;
#pragma unroll
    for (int it = 0; it < 8; ++it) {
      int mid = (lo + hi) >> 1;
      if (scsum[mid] <= p) lo = mid + 1; else hi = mid;
    }
    int idx = lo < (LL - 1) ? lo : (LL - 1);

    // ---- async copy: 2 KB row, global -> LDS -> global, B128 per lane.
    // INST_OFFSET advances both the LDS and memory addresses, so one
    // {lds,voff} pair covers all four 512 B chunks per wave.
    uint32_t lbase  = lds_byte_offset(&xbuf[wave][0]) + (uint32_t)lane * 16u;
    uint32_t srcOff = ((uint32_t)(b * LL + idx) * DD) * 4u + (uint32_t)lane * 16u;
    uint64_t attB   = (uint64_t)(uintptr_t)att;
    uint64_t outB   = (uint64_t)(uintptr_t)out;

    asm volatile(
        "global_load_async_to_lds_b128 %0, %1, %2\n\t"
        "global_load_async_to_lds_b128 %0, %1, %2 offset:512\n\t"
        "global_load_async_to_lds_b128 %0, %1, %2 offset:1024\n\t"
        "global_load_async_to_lds_b128 %0, %1, %2 offset:1536"
        :
        : "v"(lbase), "v"(srcOff), "s"(attB)
        : "memory");
    asm volatile("s_wait_asynccnt 0" ::: "memory");
    asm volatile(
        "global_store_async_from_lds_b128 %0, %1, %2 th:TH_STORE_NT\n\t"
        "global_store_async_from_lds_b128 %0, %1, %2 offset:512 th:TH_STORE_NT\n\t"
        "global_store_async_from_lds_b128 %0, %1, %2 offset:1024 th:TH_STORE_NT\n\t"
        "global_store_async_from_lds_b128 %0, %1, %2 offset:1536 th:TH_STORE_NT"
        :
        : "v"(dstOff), "v"(lbase), "s"(outB)
        : "memory");
    // S_ENDPGM performs an implicit wait-idle; no trailing asynccnt wait.
  } else {
    // masked tail: stream zeros, 4x non-temporal b128 per lane
    v4f z = (v4f)0.0f;
    v4f* dst = (v4f*)((char*)out + dstOff);
#pragma unroll
    for (int k = 0; k < 4; ++k) __builtin_nontemporal_store(z, dst + 32 * k);
  }
}

extern "C" void kernel_launch(void* const* d_in, const int* in_sizes, int n_in,
                              void* d_out, int out_size, void* d_ws, size_t ws_size,
                              hipStream_t stream) {
  (void)in_sizes; (void)n_in; (void)out_size; (void)d_ws; (void)ws_size;
  const float* att   = (const float*)d_in[0];
  const float* dur   = (const float*)d_in[1];
  const int*   alpha = (const int*)d_in[2];
  float*       out   = (float*)d_out;

  dim3 grid((BB * MAXLEN) / ROWS_PB);   // 8192 blocks
  dim3 block(256);                      // 8 wave32s
  length_regulator_kernel<<<grid, block, 0, stream>>>(att, dur, alpha, out);
}